// MDAMDecoder_45535243272801
// MI455X (gfx1250) — compile-verified
//
#include <hip/hip_runtime.h>
#include <math.h>

// ---------------- problem constants ----------------
#define B_    1024
#define N_    100
#define E_    128
#define H_    8
#define P_    5
#define DK    16
#define NEGV  (-1.0e9f)
#define CLIPC 10.0f

// ext tile columns: [gk 0:128 | gv 128:256 | lk2 256:384 | Qa 384:512 | Qb 512:640]
#define FCOLS 640
#define PSTR  641          // LDS row stride (stride%64==1 -> conflict-free columns)
#define MT    7            // ceil(100/16) row tiles
#define FT    (FCOLS/16)   // 40 column tiles
#define KSTEPS (E_/4)      // 32 wmma k-steps per tile

typedef float v2f __attribute__((ext_vector_type(2)));
typedef float v8f __attribute__((ext_vector_type(8)));

// --------------------------------------------------------------------------
// Kernel 0: per-path tiny precompute.
//   G[p]  = Wo[p]^T @ Wn_lk[p]   (folds the glimpse projection into lk)
//   q0[p] = Ws[p] @ W_placeholder (step-0 context projection)
// --------------------------------------------------------------------------
__global__ void prep_kernel(const float* __restrict__ Wn, const float* __restrict__ Wo,
                            const float* __restrict__ Ws, const float* __restrict__ Wph,
                            float* __restrict__ G, float* __restrict__ q0) {
  const int p = blockIdx.x;
  const float* wn_lk = Wn + (size_t)p * 384 * E_ + 256 * E_;   // rows 256..383 of Wn[p]
  const float* wo    = Wo + (size_t)p * E_ * E_;
  const float* wsp   = Ws + (size_t)p * E_ * 2 * E_;
  float* g = G + (size_t)p * E_ * E_;
  for (int idx = threadIdx.x; idx < E_ * E_; idx += blockDim.x) {
    const int k = idx >> 7, j = idx & 127;
    float acc = 0.f;
    for (int e = 0; e < E_; ++e) acc += wo[e * E_ + k] * wn_lk[e * E_ + j];
    g[idx] = acc;                          // G[k][j]
  }
  if (threadIdx.x < E_) {
    const int e = threadIdx.x;
    float acc = 0.f;
    for (int k = 0; k < 2 * E_; ++k) acc += wsp[e * 2 * E_ + k] * Wph[k];
    q0[p * E_ + e] = acc;
  }
}

// --------------------------------------------------------------------------
// Kernel 1: fused per-(batch,path) decoder. grid = (B, P), 256 threads (8 waves).
//   Phase A: mean/fixed-context (VALU)
//   Phase B: ext[100 x 640] = h[b] @ [Wn_gk|Wn_gv|G|Ws_a|Ws_b]^T via WMMA f32 16x16x4
//            (B-fragments register-resident; one m-tile of A staged per iteration;
//             outer loops unroll-disabled to stay under 256 VGPRs -> no spills)
//   Phase C: 100 sequential decode steps, all operands resident in LDS.
// --------------------------------------------------------------------------
__global__ __launch_bounds__(256)
void decode_kernel(const float* __restrict__ h,  const float* __restrict__ Wn,
                   const float* __restrict__ Wf, const float* __restrict__ Ws,
                   const float* __restrict__ G,  const float* __restrict__ q0,
                   float* __restrict__ out_ll, float* __restrict__ acts0,
                   float* __restrict__ lp0) {
  extern __shared__ float smem[];
  float* ext    = smem;                    // 100*641
  float* q      = ext + N_ * PSTR;         // 128
  float* fc     = q + E_;                  // 128
  float* mh     = fc + E_;                 // 128
  float* heads  = mh + E_;                 // 128
  float* compat = heads + E_;              // 8*100
  float* logits = compat + H_ * N_;        // 112
  int*   visited = (int*)(logits + 112);   // 100
  int*   ictl    = visited + N_;           // [0]=first [1]=last [2]=action
  float* fctl    = (float*)(ictl + 4);     // [0]=ll [1]=max [2]=logZ

  const int b = blockIdx.x;
  const int p = blockIdx.y;
  const int tid  = threadIdx.x;
  const int lane = tid & 31;
  const int wave = tid >> 5;

  const float* hb  = h  + (size_t)b * N_ * E_;
  const float* wn  = Wn + (size_t)p * 384 * E_;
  const float* wf  = Wf + (size_t)p * E_ * E_;
  const float* wsp = Ws + (size_t)p * E_ * 2 * E_;
  const float* gp  = G  + (size_t)p * E_ * E_;
  const float* q0p = q0 + p * E_;

  // ---- Phase A: graph mean + fixed context ----
  if (tid < E_) {
    float acc = 0.f;
    for (int n = 0; n < N_; ++n) acc += hb[n * E_ + tid];
    mh[tid] = acc * (1.0f / N_);
  }
  __syncthreads();
  if (tid < E_) {
    float acc = 0.f;
    for (int k = 0; k < E_; ++k) acc += wf[tid * E_ + k] * mh[k];
    fc[tid] = acc;
  }
  if (tid < N_) visited[tid] = 0;
  if (tid == 0) { ictl[0] = 0; ictl[1] = 0; fctl[0] = 0.f; }

  // ---- Phase B: WMMA precompute of ext = hb @ Wbig^T ----
  // A fragment (16x4 f32): lane L -> M = m0+(L&15), K-pair base = k0 + 2*(L>>4)
  // B fragment (4x16 f32): lane L -> N = f0+(L&15), same K-pair
  // D (16x16 f32, 8 VGPRs): vgpr r -> M = m0 + r + 8*(L>>4), N = f0+(L&15)
  //
  // D row m depends only on A row m, so A rows >= 100 may hold garbage
  // (address clamped to a valid row) -- those D rows are simply not stored.
  const int halfSel = lane >> 4;           // selects K pair {0,1} vs {2,3}
  const int lr = lane & 15;
#pragma clang loop unroll(disable)
  for (int ft = wave; ft < FT; ft += 8) {  // 5 f-tiles per wave
    const int f0 = ft * 16;
    const int fcol = f0 + lr;
    const float* brow;                     // row `fcol` of the fused weight matrix
    if (fcol < 256)      brow = wn  + (size_t)fcol * E_;                  // gk|gv
    else if (fcol < 384) brow = gp  + (size_t)(fcol - 256) * E_;          // lk2 via G
    else if (fcol < 512) brow = wsp + (size_t)(fcol - 384) * 2 * E_;      // Qa
    else                 brow = wsp + (size_t)(fcol - 512) * 2 * E_ + E_; // Qb
    // preload all B fragments for this f-tile (register resident, reused 7x)
    v2f bfrag[KSTEPS];
#pragma unroll
    for (int kk = 0; kk < KSTEPS; ++kk)
      bfrag[kk] = *(const v2f*)(brow + 4 * kk + 2 * halfSel);

#pragma clang loop unroll(disable)
    for (int mt = 0; mt < MT; ++mt) {
      const int m0 = mt * 16;
      const int m = m0 + lr;
      const float* arow = hb + (size_t)(m < N_ ? m : 0) * E_;  // clamped
      // stage this m-tile's A fragments (64 VGPRs), all loads in flight at once
      v2f afrag[KSTEPS];
#pragma unroll
      for (int kk = 0; kk < KSTEPS; ++kk)
        afrag[kk] = *(const v2f*)(arow + 4 * kk + 2 * halfSel);
      v8f d = {0.f, 0.f, 0.f, 0.f, 0.f, 0.f, 0.f, 0.f};
#pragma unroll
      for (int kk = 0; kk < KSTEPS; ++kk)
        d = __builtin_amdgcn_wmma_f32_16x16x4_f32(false, afrag[kk], false,
                                                  bfrag[kk], (short)0, d,
                                                  false, false);
#pragma unroll
      for (int r = 0; r < 8; ++r) {
        const int row = m0 + r + 8 * halfSel;
        if (row < N_) ext[row * PSTR + f0 + lr] = d[r];
      }
    }
  }
  __syncthreads();

  // ---- Phase C: 100 sequential decode steps, all in LDS ----
  const float inv_e = 0.088388347648318447f;  // 1/sqrt(128)
  for (int s = 0; s < N_; ++s) {
    // query: q = fc + (s==0 ? Ws@Wph : Qa[first] + Qb[last])
    if (tid < E_) {
      float val = fc[tid];
      if (s == 0) val += q0p[tid];
      else {
        const int f = ictl[0], l = ictl[1];
        val += ext[f * PSTR + 384 + tid] + ext[l * PSTR + 512 + tid];
      }
      q[tid] = val;
    }
    __syncthreads();

    // masked glimpse compatibilities: compat[head][n]
    for (int pr = tid; pr < H_ * N_; pr += 256) {
      const int hd = pr & 7;
      const int n = pr >> 3;
      float val;
      if (visited[n]) val = NEGV;
      else {
        const float* qs = q + hd * DK;
        const float* ks = ext + n * PSTR + hd * DK;      // gk
        float acc = 0.f;
#pragma unroll
        for (int d2 = 0; d2 < DK; ++d2) acc += qs[d2] * ks[d2];
        val = acc * 0.25f;                                // 1/sqrt(dk)
      }
      compat[hd * N_ + n] = val;
    }
    __syncthreads();

    // per-head softmax + heads: wave w owns head w (wave32 shfl reductions)
    {
      float* cw = compat + wave * N_;
      float mx = -3.0e38f;
      for (int n = lane; n < N_; n += 32) mx = fmaxf(mx, cw[n]);
      for (int off = 16; off > 0; off >>= 1) mx = fmaxf(mx, __shfl_xor(mx, off, 32));
      float ssum = 0.f;
      for (int n = lane; n < N_; n += 32) {
        const float e = __expf(cw[n] - mx);
        cw[n] = e;
        ssum += e;
      }
      for (int off = 16; off > 0; off >>= 1) ssum += __shfl_xor(ssum, off, 32);
      if (lane < DK) {
        const float* gvc = ext + 128 + wave * DK + lane;  // gv column
        float acc = 0.f;
        for (int n = 0; n < N_; ++n) acc += cw[n] * gvc[n * PSTR];
        heads[wave * DK + lane] = acc / ssum;
      }
    }
    __syncthreads();

    // logits[n] = tanh(heads . lk2[n] / sqrt(E)) * 10, masked
    for (int n = tid; n < 112; n += 256) {
      float v;
      if (n >= N_ || visited[n]) v = NEGV;
      else {
        const float* lk = ext + n * PSTR + 256;
        float acc = 0.f;
        for (int k = 0; k < E_; ++k) acc += heads[k] * lk[k];
        v = tanhf(acc * inv_e) * CLIPC;
      }
      logits[n] = v;
    }
    __syncthreads();

    // argmax (first-index tie-break, matches jnp.argmax) + logsumexp, wave 0
    if (wave == 0) {
      float mx = -3.0e38f;
      int idx = 0x7fffffff;
      for (int n = lane; n < N_; n += 32) {
        const float v = logits[n];
        if (v > mx || (v == mx && n < idx)) { mx = v; idx = n; }
      }
      for (int off = 16; off > 0; off >>= 1) {
        const float om = __shfl_xor(mx, off, 32);
        const int   oi = __shfl_xor(idx, off, 32);
        if (om > mx || (om == mx && oi < idx)) { mx = om; idx = oi; }
      }
      float z = 0.f;
      for (int n = lane; n < N_; n += 32) z += __expf(logits[n] - mx);
      for (int off = 16; off > 0; off >>= 1) z += __shfl_xor(z, off, 32);
      if (lane == 0) {
        const float logZ = __logf(z);
        fctl[0] += logits[idx] - mx - logZ;   // ll += lp[action]
        fctl[1] = mx;
        fctl[2] = logZ;
        ictl[2] = idx;
        if (s == 0) ictl[0] = idx;            // first
        ictl[1] = idx;                        // last
        if (p == 0) acts0[(size_t)b * N_ + s] = (float)idx;
      }
    }
    __syncthreads();
    if (s == 0 && tid < N_) {
      const float lpv = logits[tid] - fctl[1] - fctl[2];
      lp0[((size_t)p * B_ + b) * N_ + tid] = fmaxf(lpv, NEGV);
    }
    if (tid == 0) visited[ictl[2]] = 1;
    __syncthreads();
  }
  if (tid == 0) out_ll[(size_t)b * P_ + p] = fctl[0];
}

// --------------------------------------------------------------------------
// KL: deterministic two-stage reduction (no float atomics).
// --------------------------------------------------------------------------
__global__ void kl_partial(const float* __restrict__ lp0, float* __restrict__ part) {
  __shared__ float red[128];
  const int b = blockIdx.x, t = threadIdx.x;
  float acc = 0.f;
  for (int i = 0; i < P_; ++i) {
    for (int j = 0; j < P_; ++j) {
      if (i == j) continue;
      for (int n = t; n < N_; n += 128) {
        const float li = lp0[((size_t)i * B_ + b) * N_ + n];
        const float lj = lp0[((size_t)j * B_ + b) * N_ + n];
        acc += __expf(li) * (li - lj);
      }
    }
  }
  red[t] = acc;
  __syncthreads();
  for (int off = 64; off > 0; off >>= 1) {
    if (t < off) red[t] += red[t + off];
    __syncthreads();
  }
  if (t == 0) part[b] = red[0];
}

__global__ void kl_final(const float* __restrict__ part, float* __restrict__ kl) {
  if (threadIdx.x == 0 && blockIdx.x == 0) {
    float s = 0.f;
    for (int b = 0; b < B_; ++b) s += part[b];
    *kl = s / (float)(P_ * (P_ - 1) * B_);
  }
}

// --------------------------------------------------------------------------
extern "C" void kernel_launch(void* const* d_in, const int* in_sizes, int n_in,
                              void* d_out, int out_size, void* d_ws, size_t ws_size,
                              hipStream_t stream) {
  const float* h   = (const float*)d_in[0];  // [B,N,E]
  const float* Wph = (const float*)d_in[1];  // [2E]
  const float* Wn  = (const float*)d_in[2];  // [P,3E,E]
  const float* Wf  = (const float*)d_in[3];  // [P,E,E]
  const float* Ws  = (const float*)d_in[4];  // [P,E,2E]
  const float* Wo  = (const float*)d_in[5];  // [P,E,E]

  float* out     = (float*)d_out;
  float* out_ll  = out;                       // [B,P]
  float* out_kl  = out + (size_t)B_ * P_;     // scalar
  float* out_act = out_kl + 1;                // [B,N] as float

  float* wsf  = (float*)d_ws;
  float* G    = wsf;                          // P*E*E
  float* q0   = G + (size_t)P_ * E_ * E_;     // P*E
  float* lp0  = q0 + (size_t)P_ * E_;         // P*B*N
  float* part = lp0 + (size_t)P_ * B_ * N_;   // B

  prep_kernel<<<P_, 256, 0, stream>>>(Wn, Wo, Ws, Wph, G, q0);

  const size_t smem_floats =
      (size_t)N_ * PSTR + 4 * E_ + (size_t)H_ * N_ + 112 + (N_ + 4 + 4);
  dim3 grid(B_, P_);
  decode_kernel<<<grid, 256, smem_floats * sizeof(float), stream>>>(
      h, Wn, Wf, Ws, G, q0, out_ll, out_act, lp0);

  kl_partial<<<B_, 128, 0, stream>>>(lp0, part);
  kl_final<<<1, 32, 0, stream>>>(part, out_kl);
}